// HeteroGraphSAGE_4569845203257
// MI455X (gfx1250) — compile-verified
//
#include <hip/hip_runtime.h>
#include <hip/hip_bf16.h>
#include <math.h>

#define NUu 100000
#define NIi 50000
#define EUI 800000
#define EIU 800000
#define EUU 400000
#define DUd 128
#define DId 64
#define Hh  64
#define Oo  32

typedef __attribute__((ext_vector_type(2))) float v2f;
typedef __attribute__((ext_vector_type(8))) float v8f;

__device__ __forceinline__ float safef(float v) {
    if (__builtin_isnan(v)) return 0.0f;
    if (__builtin_isinf(v)) v = (v > 0.0f) ? 1.0f : -1.0f;
    return fminf(10.0f, fmaxf(-10.0f, v));
}

// Y[N,F] (+)= X[N,K] @ W[F,K]^T (+ bias)
// EP: 0=none, 1=safe, 2=safe+lrelu (scale applied before safe)
// 8 waves / block, one 16-row stripe per wave, all F columns per wave.
// W and the per-wave 16xK X tile are staged in LDS with +2-float row padding
// (stride K+2 -> lane bank offsets of 2 -> conflict-free ds_load_b64 fragments).
template<int K, int NT, int BETA, int EP, int FIXNAN>
__global__ void __launch_bounds__(256)
wmma_gemm(const float* __restrict__ X, const float* __restrict__ W,
          const float* __restrict__ bias, float* __restrict__ Y,
          int N, float scale)
{
    constexpr int F  = NT * 16;
    constexpr int KP = K + 2;          // padded LDS row stride (floats)
    constexpr int K4 = K / 4;
    __shared__ float Ws[F * KP];
    __shared__ float Xs[8 * 16 * KP];

    const int wave = threadIdx.x >> 5;
    const int lane = threadIdx.x & 31;

    // ---- stage W -> LDS (padded rows), float4-coalesced ----
    #pragma unroll
    for (int i = threadIdx.x; i < (F * K) / 4; i += 256) {
        const float4 v = ((const float4*)W)[i];
        const int r  = i / K4;
        const int c4 = i - r * K4;
        float* d = &Ws[r * KP + (c4 << 2)];
        d[0] = v.x; d[1] = v.y; d[2] = v.z; d[3] = v.w;
    }

    const int tile = blockIdx.x * 8 + wave;
    const int row0 = tile << 4;
    const bool active = (row0 < N);

    // ---- stage this wave's 16xK X tile -> LDS (padded), float4-coalesced ----
    float* xs = &Xs[wave * 16 * KP];
    if (active) {
        const float4* xg4 = (const float4*)(X + (size_t)row0 * K);
        #pragma unroll
        for (int i = lane; i < (16 * K) / 4; i += 32) {
            const float4 v = xg4[i];
            const int r  = i / K4;
            const int c4 = i - r * K4;
            float* d = &xs[r * KP + (c4 << 2)];
            d[0] = v.x; d[1] = v.y; d[2] = v.z; d[3] = v.w;
        }
    }
    __syncthreads();
    if (!active) return;

    const int half = lane >> 4;   // 0: lanes 0-15, 1: lanes 16-31
    const int l16  = lane & 15;

    // ---- A fragments: lane holds A[row0+l16][4s + 2*half + j], j=0,1 ----
    const float* xa = &xs[l16 * KP + (half << 1)];
    v2f a[K4];
    #pragma unroll
    for (int s = 0; s < K4; ++s) {
        float ax = xa[(s << 2)];
        float ay = xa[(s << 2) + 1];
        if (FIXNAN) {
            ax = __builtin_isnan(ax) ? 0.0f : ax;
            ay = __builtin_isnan(ay) ? 0.0f : ay;
        }
        a[s].x = ax; a[s].y = ay;
    }

    // ---- K loop: rotate over NT independent accumulator chains ----
    v8f acc[NT] = {};
    const float* wb = &Ws[l16 * KP + (half << 1)];
    #pragma unroll
    for (int s = 0; s < K4; ++s) {
        #pragma unroll
        for (int t = 0; t < NT; ++t) {
            const float* p = wb + (t << 4) * KP + (s << 2);
            v2f b; b.x = p[0]; b.y = p[1];
            acc[t] = __builtin_amdgcn_wmma_f32_16x16x4_f32(
                false, a[s], false, b, (short)0, acc[t], false, false);
        }
    }

    // ---- epilogue: C/D layout row = row0 + v + 8*half, col = t*16 + l16 ----
    #pragma unroll
    for (int t = 0; t < NT; ++t) {
        const int col = (t << 4) + l16;
        const float bv = bias ? bias[col] : 0.0f;
        #pragma unroll
        for (int v = 0; v < 8; ++v) {
            const int row = row0 + v + (half << 3);
            const size_t idx = (size_t)row * F + col;
            float r = acc[t][v] + bv;
            if (BETA) r += Y[idx];
            if (EP) {
                r *= scale;
                r = safef(r);
                if (EP == 2) r = (r >= 0.0f) ? r : 0.1f * r;
            }
            Y[idx] = r;
        }
    }
}

__global__ void zero_kernel(float* __restrict__ p, int n) {
    int i = blockIdx.x * blockDim.x + threadIdx.x;
    if (i < n) p[i] = 0.0f;
}

__global__ void degree_kernel(const int* __restrict__ edges, int E, float* __restrict__ cnt) {
    int e = blockIdx.x * blockDim.x + threadIdx.x;
    if (e >= E) return;
    atomicAdd(&cnt[edges[E + e]], 1.0f);
}

// one thread per (edge, feature); feature dim fixed at 64
__global__ void scatter_add_kernel(const float* __restrict__ feat, const int* __restrict__ edges,
                                   int E, float* __restrict__ agg) {
    long long tid = (long long)blockIdx.x * blockDim.x + threadIdx.x;
    if (tid >= ((long long)E << 6)) return;
    int e = (int)(tid >> 6);
    int f = (int)(tid & 63);
    int s = edges[e];
    int d = edges[E + e];
    atomicAdd(&agg[((size_t)d << 6) + f], feat[((size_t)s << 6) + f]);
}

__global__ void divide_kernel(float* __restrict__ agg, const float* __restrict__ cnt, int n_nodes) {
    int tid = blockIdx.x * blockDim.x + threadIdx.x;
    if (tid >= (n_nodes << 6)) return;
    agg[tid] /= fmaxf(cnt[tid >> 6], 1.0f);
}

// pred = sigmoid(lrelu(hu @ h1W^T + h1b) @ h2W^T + h2b)
__global__ void head_kernel(const float* __restrict__ hu, const float* __restrict__ h1W,
                            const float* __restrict__ h1b, const float* __restrict__ h2W,
                            const float* __restrict__ h2b, float* __restrict__ pred, int n) {
    int u = blockIdx.x * blockDim.x + threadIdx.x;
    if (u >= n) return;
    float x[32];
    #pragma unroll
    for (int i = 0; i < 32; ++i) x[i] = hu[((size_t)u << 5) + i];
    float z = h2b[0];
    #pragma unroll
    for (int j = 0; j < 16; ++j) {
        float s = h1b[j];
        #pragma unroll
        for (int i = 0; i < 32; ++i) s += h1W[(j << 5) + i] * x[i];
        s = (s >= 0.0f) ? s : 0.1f * s;
        z += h2W[j] * s;
    }
    pred[u] = 1.0f / (1.0f + expf(-z));
}

// runtime -> compile-time dispatch (only these 8 shapes are used)
static inline void launch_gemm(const float* X, const float* W, const float* b, float* Y,
                               int N, int K, int F, int beta, int ep, float scale, int fixnan,
                               hipStream_t stream)
{
    const int blocks = ((N >> 4) + 7) >> 3;   // 8 waves per block
    if (K == 128) {
        wmma_gemm<128, 4, 0, 1, 1><<<blocks, 256, 0, stream>>>(X, W, b, Y, N, scale);
    } else if (F == 64) {
        if (fixnan)        wmma_gemm<64, 4, 0, 1, 1><<<blocks, 256, 0, stream>>>(X, W, b, Y, N, scale);
        else if (!beta)    wmma_gemm<64, 4, 0, 0, 0><<<blocks, 256, 0, stream>>>(X, W, b, Y, N, scale);
        else if (ep == 0)  wmma_gemm<64, 4, 1, 0, 0><<<blocks, 256, 0, stream>>>(X, W, b, Y, N, scale);
        else               wmma_gemm<64, 4, 1, 2, 0><<<blocks, 256, 0, stream>>>(X, W, b, Y, N, scale);
    } else {               // F == 32
        if (!beta)         wmma_gemm<64, 2, 0, 0, 0><<<blocks, 256, 0, stream>>>(X, W, b, Y, N, scale);
        else if (ep == 0)  wmma_gemm<64, 2, 1, 0, 0><<<blocks, 256, 0, stream>>>(X, W, b, Y, N, scale);
        else               wmma_gemm<64, 2, 1, 2, 0><<<blocks, 256, 0, stream>>>(X, W, b, Y, N, scale);
    }
}

extern "C" void kernel_launch(void* const* d_in, const int* in_sizes, int n_in,
                              void* d_out, int out_size, void* d_ws, size_t ws_size,
                              hipStream_t stream)
{
    (void)in_sizes; (void)n_in; (void)out_size; (void)ws_size;

    const float* x_user  = (const float*)d_in[0];
    const float* x_item  = (const float*)d_in[1];
    const int*   e_ui    = (const int*)d_in[2];
    const int*   e_iu    = (const int*)d_in[3];
    const int*   e_uu    = (const int*)d_in[4];
    const float* pw_user = (const float*)d_in[5];
    const float* pb_user = (const float*)d_in[6];
    const float* pw_item = (const float*)d_in[7];
    const float* pb_item = (const float*)d_in[8];
    // setup_inputs dict order: per edge-type block of 6 (c1_Wl, c1_bl, c1_Wr, c2_Wl, c2_bl, c2_Wr)
    const float* c1_ui_Wl = (const float*)d_in[9];
    const float* c1_ui_bl = (const float*)d_in[10];
    const float* c1_ui_Wr = (const float*)d_in[11];
    const float* c2_ui_Wl = (const float*)d_in[12];
    const float* c2_ui_bl = (const float*)d_in[13];
    const float* c2_ui_Wr = (const float*)d_in[14];
    const float* c1_iu_Wl = (const float*)d_in[15];
    const float* c1_iu_bl = (const float*)d_in[16];
    const float* c1_iu_Wr = (const float*)d_in[17];
    const float* c2_iu_Wl = (const float*)d_in[18];
    const float* c2_iu_bl = (const float*)d_in[19];
    const float* c2_iu_Wr = (const float*)d_in[20];
    const float* c1_uu_Wl = (const float*)d_in[21];
    const float* c1_uu_bl = (const float*)d_in[22];
    const float* c1_uu_Wr = (const float*)d_in[23];
    const float* c2_uu_Wl = (const float*)d_in[24];
    const float* c2_uu_bl = (const float*)d_in[25];
    const float* c2_uu_Wr = (const float*)d_in[26];
    const float* h1_W = (const float*)d_in[27];
    const float* h1_b = (const float*)d_in[28];
    const float* h2_W = (const float*)d_in[29];
    const float* h2_b = (const float*)d_in[30];

    float* ws = (float*)d_ws;
    float* HU0  = ws; ws += (size_t)NUu * Hh;
    float* HI0  = ws; ws += (size_t)NIi * Hh;
    float* HU1  = ws; ws += (size_t)NUu * Hh;
    float* HI1  = ws; ws += (size_t)NIi * Hh;
    float* AGGU = ws; ws += (size_t)NUu * Hh;
    float* AGGI = ws; ws += (size_t)NIi * Hh;
    float* cnt_ui = ws; ws += NIi;   // item in-degree via e_ui
    float* cnt_iu = ws; ws += NUu;   // user in-degree via e_iu
    float* cnt_uu = ws; ws += NUu;   // user in-degree via e_uu

    float* pred = (float*)d_out;
    float* HU2  = pred + NUu;                  // final hu [NU,32]
    float* HI2  = HU2 + (size_t)NUu * Oo;      // final hi [NI,32]

    auto zero = [&](float* p, int n) {
        zero_kernel<<<(n + 255) / 256, 256, 0, stream>>>(p, n);
    };
    auto gemm = [&](const float* X, const float* W, const float* b, float* Y,
                    int N, int K, int F, int beta, int ep, float scale, int fixnan) {
        launch_gemm(X, W, b, Y, N, K, F, beta, ep, scale, fixnan, stream);
    };
    auto scatter = [&](const float* feat, const int* edges, int E, float* agg) {
        long long tot = (long long)E << 6;
        int blocks = (int)((tot + 255) / 256);
        scatter_add_kernel<<<blocks, 256, 0, stream>>>(feat, edges, E, agg);
    };
    auto divide = [&](float* agg, const float* cnt, int n_nodes) {
        int n = n_nodes << 6;
        divide_kernel<<<(n + 255) / 256, 256, 0, stream>>>(agg, cnt, n_nodes);
    };

    // ---- degrees (counts contiguous: cnt_ui | cnt_iu | cnt_uu) ----
    zero(cnt_ui, NIi + 2 * NUu);
    degree_kernel<<<(EUI + 255) / 256, 256, 0, stream>>>(e_ui, EUI, cnt_ui);
    degree_kernel<<<(EIU + 255) / 256, 256, 0, stream>>>(e_iu, EIU, cnt_iu);
    degree_kernel<<<(EUU + 255) / 256, 256, 0, stream>>>(e_uu, EUU, cnt_uu);

    // ---- input projections: H0 = safe(nan_to_num(x) @ pw^T + pb) ----
    gemm(x_user, pw_user, pb_user, HU0, NUu, DUd, Hh, 0, 1, 1.0f, 1);
    gemm(x_item, pw_item, pb_item, HI0, NIi, DId, Hh, 0, 1, 1.0f, 1);

    // ---- one hetero SAGE layer ----
    auto conv = [&](const float* Wl_ui, const float* bl_ui, const float* Wr_ui,
                    const float* Wl_iu, const float* bl_iu, const float* Wr_iu,
                    const float* Wl_uu, const float* bl_uu, const float* Wr_uu,
                    const float* HUin, const float* HIin,
                    float* HUout, float* HIout, int F) {
        // dst = item (user->item): hi' = lrelu(safe(agg@Wl^T + bl + hi@Wr^T))
        zero(AGGI, NIi * Hh);
        scatter(HUin, e_ui, EUI, AGGI);
        divide(AGGI, cnt_ui, NIi);
        gemm(AGGI, Wl_ui, bl_ui, HIout, NIi, Hh, F, 0, 0, 1.0f, 0);
        gemm(HIin, Wr_ui, nullptr, HIout, NIi, Hh, F, 1, 2, 1.0f, 0);
        // dst = user: hu' = lrelu(safe(0.5 * (sage_iu + sage_uu)))
        zero(AGGU, NUu * Hh);
        scatter(HIin, e_iu, EIU, AGGU);
        divide(AGGU, cnt_iu, NUu);
        gemm(AGGU, Wl_iu, bl_iu, HUout, NUu, Hh, F, 0, 0, 1.0f, 0);
        gemm(HUin, Wr_iu, nullptr, HUout, NUu, Hh, F, 1, 0, 1.0f, 0);
        zero(AGGU, NUu * Hh);
        scatter(HUin, e_uu, EUU, AGGU);
        divide(AGGU, cnt_uu, NUu);
        gemm(AGGU, Wl_uu, bl_uu, HUout, NUu, Hh, F, 1, 0, 1.0f, 0);
        gemm(HUin, Wr_uu, nullptr, HUout, NUu, Hh, F, 1, 2, 0.5f, 0);
    };

    conv(c1_ui_Wl, c1_ui_bl, c1_ui_Wr,
         c1_iu_Wl, c1_iu_bl, c1_iu_Wr,
         c1_uu_Wl, c1_uu_bl, c1_uu_Wr,
         HU0, HI0, HU1, HI1, Hh);

    conv(c2_ui_Wl, c2_ui_bl, c2_ui_Wr,
         c2_iu_Wl, c2_iu_bl, c2_iu_Wr,
         c2_uu_Wl, c2_uu_bl, c2_uu_Wr,
         HU1, HI1, HU2, HI2, Oo);

    // ---- prediction head ----
    head_kernel<<<(NUu + 255) / 256, 256, 0, stream>>>(HU2, h1_W, h1_b, h2_W, h2_b, pred, NUu);
}